// BatchedBSplines_34402688041296
// MI455X (gfx1250) — compile-verified
//
#include <hip/hip_runtime.h>

typedef __attribute__((ext_vector_type(2))) float v2f;
typedef __attribute__((ext_vector_type(8))) float v8f;

#define NB      8
#define IN_DIM  128
#define OUT_DIM 128
#define NEVAL   256
#define NCPS    16
#define KPAD    20   // 18 padded-cp columns (16 cp + 2 repeats of last) + 2 zero pad -> multiple of 4

// B-spline degree k=3, m=15, uniform knots t[idx] = (idx-3)/14.
// out[b,i,o,e] = sum_j w_j(b,i,e) * padded_cp[i,o,left-3+j]  -> per-(b,i) GEMM:
//   Acp (128 x 20)  @  W (20 x 256)  =  out tile (128 x 256)
// done with V_WMMA_F32_16X16X4_F32 (full fp32 precision, K in steps of 4).

__global__ __launch_bounds__(256)
void BatchedBSplines_wmma_kernel(const float* __restrict__ x,
                                 const float* __restrict__ cp,
                                 float* __restrict__ out) {
    // LDS: A tile (row-major, stride KPAD) and weight matrix stored K-pair-interleaved:
    //   Wp[(krow>>1)][e][krow&1]  so each lane reads its two B K-values as one 8B load.
    __shared__ float Acp[OUT_DIM * KPAD];            // 10240 B
    __shared__ float Wp[(KPAD / 2) * NEVAL * 2];     // 20480 B

    const int tid  = threadIdx.x;
    const int bi   = blockIdx.x;          // 0 .. NB*IN_DIM-1
    const int b    = bi >> 7;
    const int i    = bi & (IN_DIM - 1);

    // ---- build padded control-point tile Acp (128 x 20) ----
    const float* cpi = cp + (size_t)i * OUT_DIM * NCPS;
    for (int idx = tid; idx < OUT_DIM * KPAD; idx += 256) {
        int o = idx / KPAD;
        int c = idx - o * KPAD;
        float v;
        if (c < NCPS)    v = cpi[o * NCPS + c];
        else if (c < 18) v = cpi[o * NCPS + (NCPS - 1)];   // repeat last cp (k-1 times)
        else             v = 0.0f;                          // K padding
        Acp[idx] = v;
    }

    // ---- zero the weight matrix ----
    for (int idx = tid; idx < (KPAD / 2) * NEVAL * 2; idx += 256) Wp[idx] = 0.0f;
    __syncthreads();

    // ---- de Boor basis weights: one eval point per thread ----
    {
        const int e = tid;
        float xv = x[((size_t)b * IN_DIM + i) * NEVAL + e];
        float xc = fminf(fmaxf(xv, 0.0f), 1.0f);
        int left = (int)floorf(14.0f * xc) + 3;           // in [3, 17]
        // alpha_{r,j} = (14x - (j + left - 6)) / (4 - r)  (unclamped x, as in reference)
        float base = 14.0f * xv - (float)left + 6.0f;

        // run recursion on basis vectors: d[j][c] = coeff of gathered cp #c in d_j
        float d[4][4];
        #pragma unroll
        for (int a = 0; a < 4; ++a)
            #pragma unroll
            for (int c = 0; c < 4; ++c) d[a][c] = (a == c) ? 1.0f : 0.0f;

        #pragma unroll
        for (int r = 1; r <= 3; ++r) {
            float inv = 1.0f / (float)(4 - r);
            #pragma unroll
            for (int j = 3; j >= 1; --j) {
                if (j < r) continue;
                float alpha = (base - (float)j) * inv;
                #pragma unroll
                for (int c = 0; c < 4; ++c)
                    d[j][c] = (1.0f - alpha) * d[j - 1][c] + alpha * d[j][c];
            }
        }
        // scatter 4 weights into column e of W (rows left-3 .. left, all in [0,17])
        #pragma unroll
        for (int c = 0; c < 4; ++c) {
            int krow = left - 3 + c;
            Wp[(krow >> 1) * (NEVAL * 2) + e * 2 + (krow & 1)] = d[3][c];
        }
    }
    __syncthreads();

    // ---- GEMM via V_WMMA_F32_16X16X4_F32 ----
    const int lane = tid & 31;
    const int w    = tid >> 5;                 // wave id = M tile (8 tiles of 16)
    const int mrow = w * 16 + (lane & 15);     // A: lane -> M row
    const int kb   = (lane >> 4) * 2;          // A/B: lane half -> K pair {0,1} or {2,3}

    // hoist A fragments for all 5 K-steps (invariant across N tiles)
    v2f afrag[5];
    #pragma unroll
    for (int s = 0; s < 5; ++s) {
        int kcol = 4 * s + kb;
        afrag[s].x = Acp[mrow * KPAD + kcol];
        afrag[s].y = Acp[mrow * KPAD + kcol + 1];
    }

    float* outBase = out + ((size_t)b * IN_DIM + i) * (size_t)OUT_DIM * NEVAL;
    const int ecol = (lane & 15);

    for (int n = 0; n < 16; ++n) {             // 16 N tiles of 16 eval points
        v8f acc = {};
        #pragma unroll
        for (int s = 0; s < 5; ++s) {
            int pair = 2 * s + (lane >> 4);
            v2f bfrag;
            bfrag.x = Wp[pair * (NEVAL * 2) + (n * 16 + ecol) * 2 + 0];
            bfrag.y = Wp[pair * (NEVAL * 2) + (n * 16 + ecol) * 2 + 1];
            acc = __builtin_amdgcn_wmma_f32_16x16x4_f32(
                false, afrag[s], false, bfrag, (short)0, acc, false, false);
        }
        // D layout: VGPR v -> M = v + 8*(lane>>4), N = lane&15
        int obase = w * 16 + ((lane >> 4) << 3);
        int ee    = n * 16 + ecol;
        #pragma unroll
        for (int v = 0; v < 8; ++v) {
            outBase[(size_t)(obase + v) * NEVAL + ee] = acc[v];
        }
    }
}

extern "C" void kernel_launch(void* const* d_in, const int* in_sizes, int n_in,
                              void* d_out, int out_size, void* d_ws, size_t ws_size,
                              hipStream_t stream) {
    const float* x  = (const float*)d_in[0];   // (8, 128, 256)
    const float* cp = (const float*)d_in[1];   // (128, 128, 16)
    // d_in[2] is k==3 (compile-time constant here)
    float* out = (float*)d_out;                // (8, 128, 128, 256)

    dim3 grid(NB * IN_DIM);   // 1024 workgroups, one per (b, inDim) pair
    dim3 block(256);          // 8 waves: one M-tile each
    BatchedBSplines_wmma_kernel<<<grid, block, 0, stream>>>(x, cp, out);
}